// BaseMean_add_mp1_12910671692355
// MI455X (gfx1250) — compile-verified
//
#include <hip/hip_runtime.h>

typedef __attribute__((ext_vector_type(2))) float v2f;
typedef __attribute__((ext_vector_type(4))) float v4f;
typedef __attribute__((ext_vector_type(8))) float v8f;

#define B_TOT 8192
#define N1    201
#define NI    200
#define DD    64
#define NEG_SLOPE 0.01f

// ws layout (floats): [0,4096) W12[e][d] = sum_k W2[e,k]*W1[k,d]
//                     [4096,4160) cvec[e] = b2[e] + sum_k W2[e,k]*b1[k]
__global__ __launch_bounds__(256) void precompute_kernel(
    const float* __restrict__ w1, const float* __restrict__ w1b,
    const float* __restrict__ w2, const float* __restrict__ w2b,
    float* __restrict__ ws) {
  const int t = threadIdx.x;
  for (int j = 0; j < 16; ++j) {
    const int idx = t + 256 * j;
    const int e = idx >> 6, d = idx & 63;
    float s = 0.f;
#pragma unroll 8
    for (int k = 0; k < DD; ++k) s += w2[e * DD + k] * w1[k * DD + d];
    ws[idx] = s;
  }
  if (t < DD) {
    float s = w2b[t];
#pragma unroll 8
    for (int k = 0; k < DD; ++k) s += w2[t * DD + k] * w1b[k];
    ws[4096 + t] = s;
  }
}

// out[n,e] = lrelu( sum_d i[n,d]*Wb[e,d] + c[e] ),
// Wb[e,d] = W2[e,d] + mu[d]*W12[e,d] + (e==d)*u[e]   (diagonal folds in the u*i term)
__global__ __launch_bounds__(256) void fused_kernel(
    const float* __restrict__ emb, const float* __restrict__ w2,
    const float* __restrict__ ws, float* __restrict__ out) {
  __shared__ float mu_s[DD];
  __shared__ float u_s[DD];
  __shared__ float cv_s[DD];
  __shared__ float part[16][DD];         // 4 KB mean partials
  __shared__ float Wb[DD * 68];          // padded stride 68: conflict-free ds_load_b64

  const int t = threadIdx.x;
  const int b = blockIdx.x;
  const float* ebase = emb + (size_t)b * (N1 * DD);
  const float* irow  = ebase + DD;       // i_em rows

  // ---- phase 0: column sums of i_em via coalesced b128 sweeps (16 rows/iter).
  //      Regular-temporal loads: lines stay L2/WGP$-resident for the phase-2 reread.
  {
    const int cg = (t & 15) * 4;         // 4-col group
    const int rg = t >> 4;               // row group 0..15
    v4f sum4 = {};
    for (int r = rg; r < NI; r += 16)
      sum4 += *(const v4f*)(irow + r * DD + cg);
    *(v4f*)&part[rg][cg] = sum4;
  }
  __syncthreads();
  if (t < DD) {
    float s = 0.f;
#pragma unroll
    for (int g = 0; g < 16; ++g) s += part[g][t];
    mu_s[t] = s * (1.0f / NI);
    u_s[t]  = ebase[t];
    cv_s[t] = ws[4096 + t];
  }
  __syncthreads();

  // ---- phase 1: fused per-batch weight (b128), diagonal u-term folded in ----
#pragma unroll
  for (int j = 0; j < 4; ++j) {
    const int idx = (t + 256 * j) * 4;
    const int e = idx >> 6, d = idx & 63;
    v4f wv  = *(const v4f*)(w2 + idx);
    v4f w12 = *(const v4f*)(ws + idx);
    v4f muv = *(const v4f*)(mu_s + d);
    v4f r = wv + muv * w12;
    const float ue = u_s[e];
#pragma unroll
    for (int k = 0; k < 4; ++k) r[k] += (d + k == e) ? ue : 0.0f;
    *(v4f*)&Wb[e * 68 + d] = r;
  }
  __syncthreads();

  // ---- phase 2: WMMA fp32 GEMM ----
  const int wave  = t >> 5;
  const int lane  = t & 31;
  const int mrow  = lane & 15;           // M (A) / N (B) index within tile
  const int khalf = (lane >> 4) << 1;    // K half-select: 0 or 2
  const int mhi   = (lane >> 4) << 3;    // C/D row offset: 0 or 8

  for (int tile = wave; tile < 13; tile += 8) {   // 13 row-tiles of 16
    const int n0 = tile * 16;
    int arow = n0 + mrow;
    if (arow > NI - 1) arow = NI - 1;    // clamp tail rows (masked at store)
    const float* ap = irow + arow * DD + khalf;

    v2f afrag[16];                       // 16x64 A tile, reused across 4 e-tiles
#pragma unroll
    for (int s = 0; s < 16; ++s)         // NT: last use of these lines -> evict early
      afrag[s] = __builtin_nontemporal_load((const v2f*)(ap + 4 * s));

    const bool storeHalf = (n0 + mhi + 8 <= NI);  // whole half-wave valid (NI%16==8)

    for (int et = 0; et < 4; ++et) {
      const int e0 = et * 16;
      v8f acc = {};
      const float* bp = &Wb[(e0 + mrow) * 68 + khalf];
#pragma unroll
      for (int s = 0; s < 16; ++s) {
        v2f bfrag = *(const v2f*)(bp + 4 * s);   // B[k,n] = Wb[e0+n, 4s+k]
        acc = __builtin_amdgcn_wmma_f32_16x16x4_f32(
            false, afrag[s], false, bfrag, (short)0, acc, false, false);
      }
      // epilogue: +cvec, leaky relu, single-predicate non-temporal store
      const int e = e0 + (lane & 15);
      const float ce = cv_s[e];
      float* orow = out + (size_t)b * (NI * DD) + (size_t)(n0 + mhi) * DD + e;
      if (storeHalf) {
#pragma unroll
        for (int v = 0; v < 8; ++v) {
          float val = acc[v] + ce;
          val = (val >= 0.f) ? val : NEG_SLOPE * val;
          __builtin_nontemporal_store(val, orow + v * DD);  // write-once stream
        }
      }
    }
  }
}

extern "C" void kernel_launch(void* const* d_in, const int* in_sizes, int n_in,
                              void* d_out, int out_size, void* d_ws, size_t ws_size,
                              hipStream_t stream) {
  const float* emb = (const float*)d_in[0];
  const float* w1  = (const float*)d_in[1];
  const float* w1b = (const float*)d_in[2];
  const float* w2  = (const float*)d_in[3];
  const float* w2b = (const float*)d_in[4];
  float* out = (float*)d_out;
  float* ws  = (float*)d_ws;

  precompute_kernel<<<1, 256, 0, stream>>>(w1, w1b, w2, w2b, ws);
  fused_kernel<<<B_TOT, 256, 0, stream>>>(emb, w2, ws, out);
}